// AdvancedGNNLinkPredictor_77352361001295
// MI455X (gfx1250) — compile-verified
//
#include <hip/hip_runtime.h>

// Sizes fixed by the reference problem.
#define NN 100000      // nodes
#define NE 1250000     // edges
#define NL 200000      // label edges
#define D0 128         // input feature dim
#define HD 64          // hidden dim
#define BN_EPS 1e-5f

typedef __attribute__((ext_vector_type(2))) float v2f;
typedef __attribute__((ext_vector_type(8))) float v8f;

static __device__ __forceinline__ v8f wmma_f32x4(v2f a, v2f b, v8f c) {
  // D = A(16x4) * B(4x16) + C(16x16), dense f32 WMMA, wave32.
  return __builtin_amdgcn_wmma_f32_16x16x4_f32(false, a, false, b, (short)0, c,
                                               false, false);
}

// ---------------------------------------------------------------------------
// Degree / normalization
// ---------------------------------------------------------------------------
__global__ void k_init_deg(float* __restrict__ deg) {
  int i = blockIdx.x * 256 + threadIdx.x;
  if (i < NN) deg[i] = 1.0f;  // self-loop contributes 1 to every node's degree
}

__global__ void k_deg_edges(const int* __restrict__ ei, float* __restrict__ deg) {
  int e = blockIdx.x * 256 + threadIdx.x;
  if (e < NE) atomicAdd(&deg[ei[NE + e]], 1.0f);  // dst row of edge_index
}

__global__ void k_make_dinv(float* __restrict__ deg) {
  int i = blockIdx.x * 256 + threadIdx.x;
  if (i < NN) deg[i] = rsqrtf(deg[i]);
}

// ---------------------------------------------------------------------------
// GEMM: T[N,64] = A[N,K] @ W[K,64]   (K = 128 or 64)
// One block = 16 output rows; 4 waves each produce a 16x16 column tile.
// A tile staged in LDS with padded stride K+4 -> bank pattern (4m+k)%64.
// ---------------------------------------------------------------------------
template <int K>
__global__ __launch_bounds__(128) void k_gemm(const float* __restrict__ A,
                                              const float* __restrict__ W,
                                              float* __restrict__ T) {
  constexpr int LS = K + 4;
  __shared__ float a_lds[16 * LS];
  const int tid  = threadIdx.x;
  const int lane = tid & 31;
  const int wave = tid >> 5;
  const int row0 = blockIdx.x * 16;  // NN % 16 == 0, no guard needed

  for (int idx = tid; idx < 16 * K; idx += 128) {
    int r = idx / K, c = idx % K;  // K is a power of two -> shifts
    a_lds[r * LS + c] = A[(size_t)(row0 + r) * K + c];
  }
  __syncthreads();

  const int m   = lane & 15;   // A row / C column within tile
  const int kh  = lane >> 4;   // K half-select per ISA A/B layout
  const int col = wave * 16 + m;

  v8f c = {};
#pragma unroll
  for (int k = 0; k < K; k += 4) {
    v2f a, b;
    a.x = a_lds[m * LS + k + 2 * kh];
    a.y = a_lds[m * LS + k + 2 * kh + 1];
    b.x = W[(k + 2 * kh) * HD + col];
    b.y = W[(k + 2 * kh + 1) * HD + col];
    c = wmma_f32x4(a, b, c);
  }
#pragma unroll
  for (int i = 0; i < 8; ++i) {
    // C/D layout: VGPR i holds M = i (lanes 0-15) or M = 8+i (lanes 16-31)
    T[(size_t)(row0 + kh * 8 + i) * HD + col] = c[i];
  }
}

// agg[i,:] = T[i,:] * dinv[i]^2  (self-loop message initializes accumulator)
__global__ void k_agg_self(const float* __restrict__ T,
                           const float* __restrict__ dinv,
                           float* __restrict__ agg) {
  int idx = blockIdx.x * 256 + threadIdx.x;
  if (idx < NN * HD) {
    int i = idx >> 6;
    float di = dinv[i];
    agg[idx] = T[idx] * di * di;
  }
}

// agg[dst,:] += T[src,:] * dinv[src]*dinv[dst]; 16 lanes per edge, float4 each.
__global__ void k_scatter(const int* __restrict__ ei,
                          const float* __restrict__ T,
                          const float* __restrict__ dinv,
                          float* __restrict__ agg) {
  long long gid = (long long)blockIdx.x * 256 + threadIdx.x;
  int e = (int)(gid >> 4);
  if (e >= NE) return;
  int f = (int)(gid & 15) * 4;
  int s = ei[e];
  int d = ei[NE + e];
  float w = dinv[s] * dinv[d];
  const float4 v = *(const float4*)(T + (size_t)s * HD + f);
  float* p = agg + (size_t)d * HD + f;
  atomicAdd(p + 0, v.x * w);
  atomicAdd(p + 1, v.y * w);
  atomicAdd(p + 2, v.z * w);
  atomicAdd(p + 3, v.w * w);
}

// h = BN(agg + b) [optionally ReLU], BN folded to y = t*alpha + gamma
__global__ void k_bn_post(const float* __restrict__ agg,
                          const float* __restrict__ b, const float* __restrict__ g,
                          const float* __restrict__ be, const float* __restrict__ mn,
                          const float* __restrict__ vr, float* __restrict__ hout,
                          int relu) {
  int idx = blockIdx.x * 256 + threadIdx.x;
  if (idx >= NN * HD) return;
  int f = idx & 63;
  float alpha = g[f] * rsqrtf(vr[f] + BN_EPS);
  float gamma = (b[f] - mn[f]) * alpha + be[f];
  float y = agg[idx] * alpha + gamma;
  hout[idx] = relu ? fmaxf(y, 0.0f) : y;
}

// ---------------------------------------------------------------------------
// Decoder: per wave 16 label-edges; chained WMMA layers through padded LDS.
// ---------------------------------------------------------------------------
template <int KD, int ND, int STR, bool RELU, bool BN>
static __device__ __forceinline__ void dec_layer(
    const float* __restrict__ Z,  // LDS, wave-local base (16 rows, stride STR)
    float* __restrict__ O,        // LDS, wave-local base
    const float* __restrict__ W, const float* __restrict__ bias,
    const float* __restrict__ g, const float* __restrict__ be,
    const float* __restrict__ mn, const float* __restrict__ vr, int lane) {
  const int m  = lane & 15;
  const int kh = lane >> 4;
#pragma unroll
  for (int t = 0; t < ND / 16; ++t) {
    const int col = t * 16 + m;
    v8f c = {};
#pragma unroll
    for (int k = 0; k < KD; k += 4) {
      v2f a, b;
      a.x = Z[m * STR + k + 2 * kh];
      a.y = Z[m * STR + k + 2 * kh + 1];
      b.x = W[(k + 2 * kh) * ND + col];
      b.y = W[(k + 2 * kh + 1) * ND + col];
      c = wmma_f32x4(a, b, c);
    }
    float alpha = 1.0f, gamma = bias[col];
    if constexpr (BN) {
      float s = g[col] * rsqrtf(vr[col] + BN_EPS);
      alpha = s;
      gamma = (bias[col] - mn[col]) * s + be[col];
    }
#pragma unroll
    for (int i = 0; i < 8; ++i) {
      float y = c[i] * alpha + gamma;
      if constexpr (RELU) y = fmaxf(y, 0.0f);
      O[(kh * 8 + i) * STR + col] = y;
    }
  }
}

__global__ __launch_bounds__(128) void k_decoder(
    const float* __restrict__ h, const int* __restrict__ eli,
    const float* __restrict__ pW1, const float* __restrict__ pb1,
    const float* __restrict__ pg1, const float* __restrict__ pbe1,
    const float* __restrict__ pm1, const float* __restrict__ pv1,
    const float* __restrict__ pW2, const float* __restrict__ pb2,
    const float* __restrict__ pg2, const float* __restrict__ pbe2,
    const float* __restrict__ pm2, const float* __restrict__ pv2,
    const float* __restrict__ pW3, const float* __restrict__ pb3,
    const float* __restrict__ pW4, const float* __restrict__ pb4,
    float* __restrict__ out) {
  constexpr int S = 132;          // padded row stride: (4m+k)%64 bank pattern
  __shared__ float bufA[64 * S];  // z (64x128), later z2 (64x64)
  __shared__ float bufB[64 * S];  // z1 (64x128), later z3 (64x32)
  const int tid = threadIdx.x, lane = tid & 31, wave = tid >> 5;
  const int e0 = blockIdx.x * 64;  // NL % 64 == 0

  // cooperative gather: z[r] = concat(h[a_r], h[b_r])
  for (int idx = tid; idx < 64 * 128; idx += 128) {
    int r = idx >> 7, c = idx & 127;
    int node = (c < 64) ? eli[e0 + r] : eli[NL + e0 + r];
    bufA[r * S + c] = h[(size_t)node * HD + (c & 63)];
  }
  __syncthreads();

  const int wb = wave * 16 * S;  // each wave owns 16 private rows from here on
  dec_layer<128, 128, S, true, true>(bufA + wb, bufB + wb, pW1, pb1, pg1, pbe1,
                                     pm1, pv1, lane);
  __syncthreads();
  dec_layer<128, 64, S, true, true>(bufB + wb, bufA + wb, pW2, pb2, pg2, pbe2,
                                    pm2, pv2, lane);
  __syncthreads();
  dec_layer<64, 32, S, true, false>(bufA + wb, bufB + wb, pW3, pb3, nullptr,
                                    nullptr, nullptr, nullptr, lane);
  __syncthreads();

  // final 32 -> 1: per-lane dot product
  if (lane < 16) {
    const float* zr = bufB + wb + lane * S;
    float acc = pb4[0];
#pragma unroll
    for (int k = 0; k < 32; ++k) acc += zr[k] * pW4[k];
    out[e0 + wave * 16 + lane] = acc;
  }
}

// ---------------------------------------------------------------------------
// Host-side orchestration
// ---------------------------------------------------------------------------
extern "C" void kernel_launch(void* const* d_in, const int* in_sizes, int n_in,
                              void* d_out, int out_size, void* d_ws,
                              size_t ws_size, hipStream_t stream) {
  (void)in_sizes; (void)n_in; (void)out_size; (void)ws_size;
  const float* x   = (const float*)d_in[0];
  const int*   ei  = (const int*)d_in[1];
  const int*   eli = (const int*)d_in[2];
  const float *W0 = (const float*)d_in[3],  *b0 = (const float*)d_in[4],
              *g0 = (const float*)d_in[5],  *be0 = (const float*)d_in[6],
              *m0 = (const float*)d_in[7],  *v0 = (const float*)d_in[8];
  const float *W1 = (const float*)d_in[9],  *b1 = (const float*)d_in[10],
              *g1 = (const float*)d_in[11], *be1 = (const float*)d_in[12],
              *m1 = (const float*)d_in[13], *v1 = (const float*)d_in[14];
  const float *W2 = (const float*)d_in[15], *b2 = (const float*)d_in[16],
              *g2 = (const float*)d_in[17], *be2 = (const float*)d_in[18],
              *m2 = (const float*)d_in[19], *v2 = (const float*)d_in[20];
  const float *pW1 = (const float*)d_in[21], *pb1 = (const float*)d_in[22],
              *pW2 = (const float*)d_in[23], *pb2 = (const float*)d_in[24],
              *pW3 = (const float*)d_in[25], *pb3 = (const float*)d_in[26],
              *pW4 = (const float*)d_in[27], *pb4 = (const float*)d_in[28];
  const float *pg1 = (const float*)d_in[29], *pbe1 = (const float*)d_in[30],
              *pm1 = (const float*)d_in[31], *pv1 = (const float*)d_in[32];
  const float *pg2 = (const float*)d_in[33], *pbe2 = (const float*)d_in[34],
              *pm2 = (const float*)d_in[35], *pv2 = (const float*)d_in[36];
  float* out = (float*)d_out;

  // Workspace layout: dinv[N] | bufA[N*64] | bufB[N*64]  (~52 MB)
  float* dinv = (float*)d_ws;
  float* bufA = dinv + 100096;  // 128-float aligned past N
  float* bufB = bufA + (size_t)NN * HD;

  const int B = 256;
  // degrees -> dinv
  k_init_deg<<<(NN + B - 1) / B, B, 0, stream>>>(dinv);
  k_deg_edges<<<(NE + B - 1) / B, B, 0, stream>>>(ei, dinv);
  k_make_dinv<<<(NN + B - 1) / B, B, 0, stream>>>(dinv);

  const int gGemm = NN / 16;                    // 6250
  const int gScat = (NE * 16) / B;              // 78125
  const int gElem = (NN * HD) / B;              // 25000

  // layer 0: x[N,128] -> h1 in bufA
  k_gemm<D0><<<gGemm, 128, 0, stream>>>(x, W0, bufA);     // T in bufA
  k_agg_self<<<gElem, B, 0, stream>>>(bufA, dinv, bufB);  // agg in bufB
  k_scatter<<<gScat, B, 0, stream>>>(ei, bufA, dinv, bufB);
  k_bn_post<<<gElem, B, 0, stream>>>(bufB, b0, g0, be0, m0, v0, bufA, 1);

  // layer 1: h1 (bufA) -> h2 in bufB
  k_gemm<HD><<<gGemm, 128, 0, stream>>>(bufA, W1, bufB);  // T in bufB
  k_agg_self<<<gElem, B, 0, stream>>>(bufB, dinv, bufA);  // agg in bufA
  k_scatter<<<gScat, B, 0, stream>>>(ei, bufB, dinv, bufA);
  k_bn_post<<<gElem, B, 0, stream>>>(bufA, b1, g1, be1, m1, v1, bufB, 1);

  // layer 2: h2 (bufB) -> h3 in bufA (no ReLU on last GCN layer)
  k_gemm<HD><<<gGemm, 128, 0, stream>>>(bufB, W2, bufA);  // T in bufA
  k_agg_self<<<gElem, B, 0, stream>>>(bufA, dinv, bufB);  // agg in bufB
  k_scatter<<<gScat, B, 0, stream>>>(ei, bufA, dinv, bufB);
  k_bn_post<<<gElem, B, 0, stream>>>(bufB, b2, g2, be2, m2, v2, bufA, 0);

  // decoder MLP over label edges
  k_decoder<<<NL / 64, 128, 0, stream>>>(bufA, eli, pW1, pb1, pg1, pbe1, pm1,
                                         pv1, pW2, pb2, pg2, pbe2, pm2, pv2,
                                         pW3, pb3, pW4, pb4, out);
}